// GraphSAGE_77584289235644
// MI455X (gfx1250) — compile-verified
//
#include <hip/hip_runtime.h>
#include <hip/hip_bf16.h>

typedef __attribute__((ext_vector_type(2))) float v2f;
typedef __attribute__((ext_vector_type(8))) float v8f;

// Non-returning f32 atomic add straight to the global atomic unit (L2 RMW).
__device__ __forceinline__ void gatomic_add_f32(float* p, float v) {
  asm volatile("global_atomic_add_f32 %0, %1, off" : : "v"(p), "v"(v) : "memory");
}

// ---------------- degree kernels ----------------
__global__ __launch_bounds__(256) void k_degree(const int* __restrict__ dst,
                                                float* __restrict__ deg, int E) {
  int i = blockIdx.x * blockDim.x + threadIdx.x;
  if (i < E) gatomic_add_f32(&deg[dst[i]], 1.0f);
}

__global__ __launch_bounds__(256) void k_deginv(float* __restrict__ deg, int n) {
  int i = blockIdx.x * blockDim.x + threadIdx.x;
  if (i < n) deg[i] = 1.0f / fmaxf(deg[i], 1.0f);
}

// ---------------- WMMA GEMM: C[M,NC] = A[M,K] @ B[K,NC] (+ fused epilogue) ----
// One wave owns a 16-row stripe and ALL NC/16 column tiles: the A fragment is
// loaded once per k-step and reused across every column tile (8x less traffic
// on the wide matrix vs one-tile-per-wave).  V_WMMA_F32_16X16X4_F32 layouts:
//   A frag: lane = g*16 + m holds (M = row0+m, K = k0 + 2g + {0,1})  -> v2f
//   B frag: lane = g*16 + n holds (K = k0 + 2g + {0,1}, N = col0+n)  -> v2f
//   C frag: VGPR r, lane = g*16 + n holds (M = row0 + 8g + r, N = col0+n)
template <int K, int NC, bool FUSE, bool RELU>
__global__ __launch_bounds__(256) void k_sage_gemm(
    const float* __restrict__ A, const float* __restrict__ B,
    const float* __restrict__ bias, const float* __restrict__ agg,
    const float* __restrict__ deg_inv, float* __restrict__ C, int M) {
  constexpr int TN = NC / 16;
  int wave = blockIdx.x * (blockDim.x >> 5) + (threadIdx.x >> 5);
  if (wave >= (M >> 4)) return;  // wave-uniform (WMMA needs full EXEC)
  int lane = threadIdx.x & 31;
  int m = lane & 15;  // M-index for A, N-index for B/C
  int g = lane >> 4;  // lane-half selector
  int row0 = wave << 4;

  const float* ap = A + (size_t)(row0 + m) * K;
  v8f acc[TN] = {};
#pragma unroll 2
  for (int k0 = 0; k0 < K; k0 += 4) {
    int ka = k0 + (g << 1);
    v2f a;
    a.x = ap[ka];
    a.y = ap[ka + 1];
    const float* bp = B + (size_t)ka * NC + m;
#pragma unroll
    for (int t = 0; t < TN; ++t) {
      v2f b;
      b.x = bp[t * 16];
      b.y = bp[NC + t * 16];
      acc[t] = __builtin_amdgcn_wmma_f32_16x16x4_f32(false, a, false, b,
                                                     (short)0, acc[t], false,
                                                     false);
    }
  }

#pragma unroll
  for (int t = 0; t < TN; ++t) {
    int col = t * 16 + m;
#pragma unroll
    for (int r = 0; r < 8; ++r) {
      int row = row0 + (g << 3) + r;
      float v = acc[t][r];
      if constexpr (FUSE) {
        v += bias[col] + deg_inv[row] * agg[(size_t)row * NC + col];
        if constexpr (RELU) v = fmaxf(v, 0.0f);
      }
      C[(size_t)row * NC + col] = v;
    }
  }
}

// ---------------- edge scatter-add: agg[dst] += p[src] ----------------
// One wave per edge; each lane owns F/32 consecutive floats (vector load,
// then per-dword global_atomic_add_f32 into the destination row in L2).
template <int F>
__global__ __launch_bounds__(256) void k_scatter_add(
    const float* __restrict__ p, const int* __restrict__ src,
    const int* __restrict__ dst, float* __restrict__ agg, int E) {
  int e = blockIdx.x * (blockDim.x >> 5) + (threadIdx.x >> 5);
  if (e >= E) return;
  int lane = threadIdx.x & 31;
  int s = src[e], d = dst[e];
  constexpr int V = F / 32;
  const float* ps = p + (size_t)s * F + lane * V;
  float* pd = agg + (size_t)d * F + lane * V;
  float vals[V];
#pragma unroll
  for (int i = 0; i < V; ++i) vals[i] = ps[i];
#pragma unroll
  for (int i = 0; i < V; ++i) gatomic_add_f32(&pd[i], vals[i]);
}

// ---------------- launcher ----------------
extern "C" void kernel_launch(void* const* d_in, const int* in_sizes, int n_in,
                              void* d_out, int out_size, void* d_ws,
                              size_t ws_size, hipStream_t stream) {
  const float* x   = (const float*)d_in[0];
  const int*   src = (const int*)d_in[1];
  const int*   dst = (const int*)d_in[2];
  const float* W1s = (const float*)d_in[3];
  const float* W1n = (const float*)d_in[4];
  const float* b1  = (const float*)d_in[5];
  const float* W2s = (const float*)d_in[6];
  const float* W2n = (const float*)d_in[7];
  const float* b2  = (const float*)d_in[8];
  float* out = (float*)d_out;

  const int IN = 256, H = 128, OUT = 64;
  int N = in_sizes[0] / IN;  // 100000 (divisible by 16)
  int E = in_sizes[1];       // 3200000

  float* ws   = (float*)d_ws;
  float* deg  = ws;                    // [N]   -> becomes deg_inv
  float* p1   = deg + N;               // [N,H]   x @ W1_neigh
  float* agg1 = p1 + (size_t)N * H;    // [N,H]   segment-sum of p1
  float* h    = agg1 + (size_t)N * H;  // [N,H]
  float* p2   = h + (size_t)N * H;     // [N,OUT] h @ W2_neigh
  float* agg2 = p2 + (size_t)N * OUT;  // [N,OUT]

  hipMemsetAsync(deg, 0, sizeof(float) * (size_t)N, stream);
  hipMemsetAsync(agg1, 0, sizeof(float) * (size_t)N * H, stream);
  hipMemsetAsync(agg2, 0, sizeof(float) * (size_t)N * OUT, stream);

  // in-degree, then deg_inv = 1/max(deg,1) in place
  k_degree<<<(E + 255) / 256, 256, 0, stream>>>(dst, deg, E);
  k_deginv<<<(N + 255) / 256, 256, 0, stream>>>(deg, N);

  int stripes = N / 16;               // one wave per 16-row stripe
  int gemmGrid = (stripes + 7) / 8;   // 8 waves per 256-thread block

  // ---- layer 1 (project-first: halves edge traffic 256 -> 128 floats/edge)
  k_sage_gemm<256, 128, false, false><<<gemmGrid, 256, 0, stream>>>(
      x, W1n, nullptr, nullptr, nullptr, p1, N);
  k_scatter_add<128><<<(E + 7) / 8, 256, 0, stream>>>(p1, src, dst, agg1, E);
  k_sage_gemm<256, 128, true, true><<<gemmGrid, 256, 0, stream>>>(
      x, W1s, b1, agg1, deg, h, N);

  // ---- layer 2 (project-first: 128 -> 64 floats/edge)
  k_sage_gemm<128, 64, false, false><<<gemmGrid, 256, 0, stream>>>(
      h, W2n, nullptr, nullptr, nullptr, p2, N);
  k_scatter_add<64><<<(E + 7) / 8, 256, 0, stream>>>(p2, src, dst, agg2, E);
  k_sage_gemm<128, 64, true, false><<<gemmGrid, 256, 0, stream>>>(
      h, W2s, b2, agg2, deg, out, N);
}